// DiscriminativeLoss_69217692942678
// MI455X (gfx1250) — compile-verified
//
#include <hip/hip_runtime.h>
#include <math.h>

// Problem constants (match reference setup_inputs)
#define B_   8
#define D_   64
#define N_   65536      // 256*256
#define C_   32
#define PPB  1024       // points per block for the streaming passes
#define EPSF 1e-12f
#define DELTA_VAR_F  1.0f
#define DELTA_DIST_F 2.0f
#define DELTA_REG_F  8.0f   // sqrt(64)

// Workspace layout (in floats)
#define OFF_COUNTS  0
#define OFF_SUMS    (OFF_COUNTS + B_*C_)           // 256
#define OFF_CENTERS (OFF_SUMS   + B_*C_*D_)        // 16640
#define OFF_INV     (OFF_CENTERS+ B_*C_*D_)        // 33024
#define OFF_SCALEV  (OFF_INV    + B_*C_)           // 33280 : per-batch 1/(n_clusters*B)
#define OFF_ACC     (OFF_SCALEV + B_)              // 33288 : [var, dist, reg]
#define WS_FLOATS   (OFF_ACC + 3)

typedef float v2f __attribute__((ext_vector_type(2)));
typedef float v8f __attribute__((ext_vector_type(8)));

__device__ __forceinline__ void atomAddWG(float* p, float v) {
  __hip_atomic_fetch_add(p, v, __ATOMIC_RELAXED, __HIP_MEMORY_SCOPE_WORKGROUP);
}
__device__ __forceinline__ void atomAddDev(float* p, float v) {
  __hip_atomic_fetch_add(p, v, __ATOMIC_RELAXED, __HIP_MEMORY_SCOPE_AGENT);
}

// ---------------- Kernel 1: zero workspace ----------------
__global__ void k_init(float* __restrict__ ws) {
  for (int i = blockIdx.x * blockDim.x + threadIdx.x; i < WS_FLOATS;
       i += gridDim.x * blockDim.x)
    ws[i] = 0.0f;
}

// ---------------- Kernel 2: segment-sum (cluster sums + counts) ----------------
__global__ __launch_bounds__(256)
void k_accum(const float* __restrict__ x, const int* __restrict__ lab,
             float* __restrict__ ws) {
  __shared__ float lsum[C_ * 65];   // stride 65 -> bank (c+dd)%64, conflict-free
  __shared__ float lcnt[C_];
  const int tid = threadIdx.x;
  const int b   = blockIdx.y;
  const int n0  = blockIdx.x * PPB;

  for (int i = tid; i < C_ * 65; i += blockDim.x) lsum[i] = 0.0f;
  if (tid < C_) lcnt[tid] = 0.0f;
  __syncthreads();

  const float* xb = x + (size_t)b * D_ * N_;
  const int*   lb = lab + (size_t)b * N_;

  for (int p = tid; p < PPB; p += blockDim.x) {
    const int n = n0 + p;
    const int c = lb[n];
    atomAddWG(&lcnt[c], 1.0f);
    const float* xp = xb + n;              // consecutive lanes -> consecutive n
    #pragma unroll 8
    for (int dd = 0; dd < D_; ++dd)
      atomAddWG(&lsum[c * 65 + dd], xp[(size_t)dd * N_]);
  }
  __syncthreads();

  float* sums = ws + OFF_SUMS + (size_t)b * C_ * D_;
  for (int i = tid; i < C_ * D_; i += blockDim.x) {
    const int c = i >> 6, dd = i & 63;
    atomAddDev(&sums[i], lsum[c * 65 + dd]);
  }
  if (tid < C_) atomAddDev(ws + OFF_COUNTS + b * C_ + tid, lcnt[tid]);
}

// ---------------- Kernel 3: centers, WMMA Gram matrix, dist + reg terms ----------------
// One block per batch, 128 threads = 4 waves; wave w computes 16x16 tile
// (mtile=(w>>1)*16, etile=(w&1)*16) of cdot = centers * centers^T via
// V_WMMA_F32_16X16X4_F32 accumulated over K=64 in steps of 4.
__global__ __launch_bounds__(128)
void k_centers_dist_reg(float* __restrict__ ws) {
  __shared__ float sc[C_ * 65];    // centers, padded
  __shared__ float cd[C_ * 33];    // cdot, padded
  __shared__ float sinv[C_];
  __shared__ float sdum[C_];
  __shared__ float sc2[C_];
  __shared__ float snc;
  __shared__ float sred;

  const int tid = threadIdx.x;
  const int b   = blockIdx.x;
  const float* counts = ws + OFF_COUNTS + b * C_;
  const float* sums   = ws + OFF_SUMS + (size_t)b * C_ * D_;
  float* gcent        = ws + OFF_CENTERS + (size_t)b * C_ * D_;

  if (tid < C_) {
    const float cnt   = counts[tid];
    const float real  = (cnt > 0.0f) ? 1.0f : 0.0f;
    const float dummy = 1.0f - real;
    const float inv   = real / (cnt + dummy);
    sinv[tid] = inv;
    sdum[tid] = dummy;
    (ws + OFF_INV)[b * C_ + tid] = inv;
  }
  if (tid == 0) sred = 0.0f;
  __syncthreads();
  if (tid == 0) {
    float nc = 0.0f;
    for (int c = 0; c < C_; ++c) nc += 1.0f - sdum[c];
    snc = nc;
    (ws + OFF_SCALEV)[b] = 1.0f / (nc * (float)B_);
  }
  // finalize centers -> LDS + global
  for (int i = tid; i < C_ * D_; i += blockDim.x) {
    const int c = i >> 6, dd = i & 63;
    const float v = sums[i] * sinv[c];
    sc[c * 65 + dd] = v;
    gcent[i] = v;
  }
  __syncthreads();

  // WMMA Gram matrix. A (16x4 f32) layout: lanes0-15 M=0..15 K={k0,k0+1},
  // lanes16-31 K={k0+2,k0+3}. For a Gram matrix B = A^T, so the B operand
  // uses the same per-lane pattern with the column tile offset.
  {
    const int lane  = tid & 31;
    const int wave  = tid >> 5;            // 0..3
    const int mt    = (wave >> 1) * 16;
    const int et    = (wave & 1) * 16;
    const int r16   = lane & 15;
    const int khalf = (lane < 16) ? 0 : 2;
    v8f acc = {};
    #pragma unroll
    for (int k0 = 0; k0 < D_; k0 += 4) {
      v2f a, bb;
      a.x  = sc[(mt + r16) * 65 + k0 + khalf];
      a.y  = sc[(mt + r16) * 65 + k0 + khalf + 1];
      bb.x = sc[(et + r16) * 65 + k0 + khalf];
      bb.y = sc[(et + r16) * 65 + k0 + khalf + 1];
      acc = __builtin_amdgcn_wmma_f32_16x16x4_f32(
          /*neg_a=*/false, a, /*neg_b=*/false, bb,
          /*c_mod=*/(short)0, acc, /*reuse_a=*/false, /*reuse_b=*/false);
    }
    // D layout: VGPR r -> (M=mt+r, col=lane) lanes0-15 ; (M=mt+r+8) lanes16-31
    union { v8f v; float f[8]; } u; u.v = acc;
    const int rowoff = (lane < 16) ? 0 : 8;
    #pragma unroll
    for (int r = 0; r < 8; ++r)
      cd[(mt + r + rowoff) * 33 + (et + r16)] = u.f[r];
  }
  __syncthreads();

  if (tid < C_) sc2[tid] = cd[tid * 33 + tid];
  __syncthreads();

  // dist term: pairwise hinge cost (eye/dummy masking as in reference)
  float local = 0.0f;
  for (int idx = tid; idx < C_ * C_; idx += blockDim.x) {
    const int i = idx >> 5, j = idx & 31;
    const float cd2  = fmaxf(sc2[i] + sc2[j] - 2.0f * cd[i * 33 + j], EPSF);
    const float dist = sqrtf(cd2);
    const float eye  = (i == j) ? 1.0f : 0.0f;
    const float masked = dist + DELTA_DIST_F * (eye + sdum[i] + sdum[j]);
    const float t = fmaxf(DELTA_DIST_F - masked, 0.0f);
    local += t * t;
  }
  // reg term
  float lreg = 0.0f;
  if (tid < C_) {
    const float cn = sqrtf(fmaxf(sc2[tid], EPSF));
    const float t  = fmaxf(cn - DELTA_REG_F, 0.0f);
    lreg = t * t;
  }

  atomAddWG(&sred, local);
  __syncthreads();
  if (tid == 0) {
    const float nc    = snc;
    const float pairs = nc * (nc - 1.0f) * 0.5f;
    atomAddDev(ws + OFF_ACC + 1, sred / 2.0f / pairs / (float)B_);
    sred = 0.0f;
  }
  __syncthreads();
  atomAddWG(&sred, lreg);
  __syncthreads();
  if (tid == 0) atomAddDev(ws + OFF_ACC + 2, sred / snc / (float)B_);
}

// ---------------- Kernel 4: variance (pull) term ----------------
__global__ __launch_bounds__(256)
void k_var(const float* __restrict__ x, const int* __restrict__ lab,
           float* __restrict__ ws) {
  __shared__ float lc[C_ * 65];    // centers, padded
  __shared__ float linv[C_];
  __shared__ float wred[8];
  const int tid = threadIdx.x;
  const int b   = blockIdx.y;
  const int n0  = blockIdx.x * PPB;

  const float* gcent = ws + OFF_CENTERS + (size_t)b * C_ * D_;
  for (int i = tid; i < C_ * D_; i += blockDim.x)
    lc[(i >> 6) * 65 + (i & 63)] = gcent[i];
  if (tid < C_) linv[tid] = (ws + OFF_INV)[b * C_ + tid];
  __syncthreads();

  const float scale = (ws + OFF_SCALEV)[b];
  const float* xb = x + (size_t)b * D_ * N_;
  const int*   lb = lab + (size_t)b * N_;

  float acc = 0.0f;
  for (int p = tid; p < PPB; p += blockDim.x) {
    const int n = n0 + p;
    const int c = lb[n];
    const float* xp = xb + n;
    const float* cp = lc + c * 65;
    float s = 0.0f;
    #pragma unroll 8
    for (int dd = 0; dd < D_; ++dd) {
      const float diff = xp[(size_t)dd * N_] - cp[dd];
      s = fmaf(diff, diff, s);
    }
    const float dist = sqrtf(fmaxf(s, EPSF));
    const float t = fmaxf(dist - DELTA_VAR_F, 0.0f);
    acc += t * t * linv[c];
  }
  acc *= scale;

  // wave32 reduction, then cross-wave via LDS
  #pragma unroll
  for (int o = 16; o > 0; o >>= 1) acc += __shfl_down(acc, o, 32);
  if ((tid & 31) == 0) wred[tid >> 5] = acc;
  __syncthreads();
  if (tid == 0) {
    float s = 0.0f;
    #pragma unroll
    for (int w = 0; w < 8; ++w) s += wred[w];
    atomAddDev(ws + OFF_ACC + 0, s);
  }
}

// ---------------- Kernel 5: combine ----------------
__global__ void k_final(const float* __restrict__ ws, float* __restrict__ out) {
  out[0] = ws[OFF_ACC + 0] + ws[OFF_ACC + 1] + ws[OFF_ACC + 2];
}

extern "C" void kernel_launch(void* const* d_in, const int* in_sizes, int n_in,
                              void* d_out, int out_size, void* d_ws, size_t ws_size,
                              hipStream_t stream) {
  (void)in_sizes; (void)n_in; (void)out_size; (void)ws_size;
  const float* data   = (const float*)d_in[0];
  const int*   labels = (const int*)d_in[1];
  float* ws  = (float*)d_ws;
  float* out = (float*)d_out;

  k_init<<<dim3(64), dim3(256), 0, stream>>>(ws);
  k_accum<<<dim3(N_ / PPB, B_), dim3(256), 0, stream>>>(data, labels, ws);
  k_centers_dist_reg<<<dim3(B_), dim3(128), 0, stream>>>(ws);
  k_var<<<dim3(N_ / PPB, B_), dim3(256), 0, stream>>>(data, labels, ws);
  k_final<<<dim3(1), dim3(1), 0, stream>>>(ws, out);
}